// deformable_cross_attention_1322849927580
// MI455X (gfx1250) — compile-verified
//
#include <hip/hip_runtime.h>
#include <cmath>

// ---------------------------------------------------------------------------
// Problem constants (from reference)
// ---------------------------------------------------------------------------
#define EMBED 128
#define NH 8
#define NP 4
#define NL 5
#define QD 128
#define QTOT (QD * QD)          // 16384
#define BATCH 2
#define LTOT 21824              // 128^2+64^2+32^2+16^2+8^2
#define HIDDEN 512

typedef float v2f __attribute__((ext_vector_type(2)));
typedef float v8f __attribute__((ext_vector_type(8)));

static inline int cdiv(int a, int b) { return (a + b - 1) / b; }

// ---------------------------------------------------------------------------
// fp32 WMMA GEMM, register-blocked 2x2:  Y = scale * (relu?(X @ W + bias)) [+ Y]
//   X: rows x K (row major), W: K x N (row major), bias: N
//   One wave32 computes a 32x32 output macro-tile as 4 independent 16x16
//   V_WMMA_F32_16X16X4_F32 accumulator chains (hides WMMA RAW latency, and
//   doubles FLOPs per fragment byte vs a single-tile wave).
//   rows % 32 == 0 and N % 32 == 0 for all call sites.
// ---------------------------------------------------------------------------
__global__ __launch_bounds__(256) void gemm_wmma_f32(
    const float* __restrict__ X, const float* __restrict__ W,
    const float* __restrict__ bias, float* __restrict__ Y,
    int rows, int K, int N, int relu, int scaleMode,
    const float* __restrict__ alphaPtr, int accum)
{
    const int wavesPerBlock = blockDim.x >> 5;
    const int wid = blockIdx.x * wavesPerBlock + (threadIdx.x >> 5);
    const int tilesM = rows >> 5;                // 32-row macro tiles
    const int tilesN = N >> 5;                   // 32-col macro tiles
    if (wid >= tilesM * tilesN) return;          // wave-uniform: EXEC stays all-1s

    const int tm = wid % tilesM;                 // consecutive waves share B strip
    const int tn = wid / tilesM;
    const int lane = threadIdx.x & 31;
    const int mrow = lane & 15;                  // M for A frag / N for B,C,D frags
    const int kh   = lane >> 4;                  // upper half of wave holds K={2,3}

    const float* __restrict__ xrow0 = X + (size_t)(tm * 32 + mrow) * K;
    const float* __restrict__ xrow1 = xrow0 + (size_t)16 * K;
    const float* __restrict__ wcol0 = W + tn * 32 + mrow;
    const float* __restrict__ wcol1 = wcol0 + 16;

    v8f c00 = {}, c01 = {}, c10 = {}, c11 = {};
    const int ksteps = K >> 2;
    #pragma unroll 2
    for (int ks = 0; ks < ksteps; ++ks) {
        const int k0 = (ks << 2) + (kh << 1);
        v2f a0; a0.x = xrow0[k0];               a0.y = xrow0[k0 + 1];
        v2f a1; a1.x = xrow1[k0];               a1.y = xrow1[k0 + 1];
        v2f b0; b0.x = wcol0[(size_t)k0 * N];   b0.y = wcol0[(size_t)(k0 + 1) * N];
        v2f b1; b1.x = wcol1[(size_t)k0 * N];   b1.y = wcol1[(size_t)(k0 + 1) * N];
        c00 = __builtin_amdgcn_wmma_f32_16x16x4_f32(false, a0, false, b0, (short)0, c00, false, false);
        c01 = __builtin_amdgcn_wmma_f32_16x16x4_f32(false, a0, false, b1, (short)0, c01, false, false);
        c10 = __builtin_amdgcn_wmma_f32_16x16x4_f32(false, a1, false, b0, (short)0, c10, false, false);
        c11 = __builtin_amdgcn_wmma_f32_16x16x4_f32(false, a1, false, b1, (short)0, c11, false, false);
    }

    float s = 1.0f;
    if (scaleMode == 1)      s = *alphaPtr;
    else if (scaleMode == 2) s = 1.0f - *alphaPtr;

    const int col0 = tn * 32 + mrow;
    const int col1 = col0 + 16;
    const float bv0 = bias ? bias[col0] : 0.0f;
    const float bv1 = bias ? bias[col1] : 0.0f;

    #pragma unroll
    for (int r = 0; r < 8; ++r) {
        const int row0 = tm * 32 + r + (kh << 3);  // C/D layout: +8 for upper lanes
        const int row1 = row0 + 16;
        float v00 = c00[r] + bv0, v01 = c01[r] + bv1;
        float v10 = c10[r] + bv0, v11 = c11[r] + bv1;
        if (relu) {
            v00 = fmaxf(v00, 0.0f); v01 = fmaxf(v01, 0.0f);
            v10 = fmaxf(v10, 0.0f); v11 = fmaxf(v11, 0.0f);
        }
        v00 *= s; v01 *= s; v10 *= s; v11 *= s;
        if (accum) {
            v00 += Y[(size_t)row0 * N + col0]; v01 += Y[(size_t)row0 * N + col1];
            v10 += Y[(size_t)row1 * N + col0]; v11 += Y[(size_t)row1 * N + col1];
        }
        Y[(size_t)row0 * N + col0] = v00; Y[(size_t)row0 * N + col1] = v01;
        Y[(size_t)row1 * N + col0] = v10; Y[(size_t)row1 * N + col1] = v11;
    }
}

// ---------------------------------------------------------------------------
// 1x1 conv: flat[b, start+p, o] = sum_c f[b,c,p]*w[o,c] + bias[o]
// o is the fastest thread index -> coalesced 512B writes, uniform f loads,
// w table (<=160KB) lives in cache.
// ---------------------------------------------------------------------------
__global__ void conv1x1_kernel(const float* __restrict__ f, const float* __restrict__ w,
                               const float* __restrict__ bias, float* __restrict__ flat,
                               int Cin, int HW, int logHW, int start)
{
    const int t = blockIdx.x * blockDim.x + threadIdx.x;
    const int total = BATCH * HW * EMBED;
    if (t >= total) return;
    const int o = t & (EMBED - 1);
    const int p = (t >> 7) & (HW - 1);
    const int b = t >> (7 + logHW);
    const float* __restrict__ fp = f + (size_t)b * Cin * HW + p;
    const float* __restrict__ wp = w + (size_t)o * Cin;
    float acc = bias[o];
    for (int ci = 0; ci < Cin; ++ci) acc += fp[(size_t)ci * HW] * wp[ci];
    flat[((size_t)b * LTOT + start + p) * EMBED + o] = acc;
}

// GroupNorm stats: one block per (b, group); group = 4 channels x HW
__global__ __launch_bounds__(256) void gn_stats_kernel(
    const float* __restrict__ flat, float* __restrict__ stats, int HW, int start)
{
    const int b = blockIdx.x >> 5;
    const int g = blockIdx.x & 31;
    const int n = 4 * HW;
    float s = 0.0f, s2 = 0.0f;
    for (int e = threadIdx.x; e < n; e += blockDim.x) {
        const int o = (g << 2) + (e & 3);
        const int p = e >> 2;
        const float v = flat[((size_t)b * LTOT + start + p) * EMBED + o];
        s += v; s2 += v * v;
    }
    __shared__ float sh0[256], sh1[256];
    sh0[threadIdx.x] = s; sh1[threadIdx.x] = s2;
    __syncthreads();
    for (int off = 128; off > 0; off >>= 1) {
        if ((int)threadIdx.x < off) {
            sh0[threadIdx.x] += sh0[threadIdx.x + off];
            sh1[threadIdx.x] += sh1[threadIdx.x + off];
        }
        __syncthreads();
    }
    if (threadIdx.x == 0) {
        const float inv = 1.0f / (float)n;
        const float mu = sh0[0] * inv;
        const float var = sh1[0] * inv - mu * mu;
        stats[blockIdx.x * 2]     = mu;
        stats[blockIdx.x * 2 + 1] = rsqrtf(var + 1e-5f);
    }
}

__global__ void gn_apply_kernel(float* __restrict__ flat, const float* __restrict__ stats,
                                const float* __restrict__ gamma, const float* __restrict__ beta,
                                int HW, int logHW, int start)
{
    const int t = blockIdx.x * blockDim.x + threadIdx.x;
    const int total = BATCH * HW * EMBED;
    if (t >= total) return;
    const int o = t & (EMBED - 1);
    const int p = (t >> 7) & (HW - 1);
    const int b = t >> (7 + logHW);
    const int g = o >> 2;
    const float mu = stats[(b * 32 + g) * 2];
    const float rs = stats[(b * 32 + g) * 2 + 1];
    const size_t idx = ((size_t)b * LTOT + start + p) * EMBED + o;
    flat[idx] = (flat[idx] - mu) * rs * gamma[o] + beta[o];
}

__global__ void add_vec_kernel(const float* __restrict__ a, const float* __restrict__ b,
                               float* __restrict__ o, int n)
{
    const int t = blockIdx.x * blockDim.x + threadIdx.x;
    if (t < n) o[t] = a[t] + b[t];
}

// Softmax over the 20 (NL*NP) attention logits per (q, head); in place on (Q,160)
__global__ void softmax20_kernel(float* __restrict__ aw)
{
    const int t = blockIdx.x * blockDim.x + threadIdx.x;
    if (t >= QTOT * NH) return;
    const int qi = t >> 3, h = t & 7;
    float* __restrict__ p = aw + (size_t)qi * (NH * NL * NP) + h * (NL * NP);
    float m = -1e30f;
    #pragma unroll
    for (int i = 0; i < 20; ++i) m = fmaxf(m, p[i]);
    float e[20]; float s = 0.0f;
    #pragma unroll
    for (int i = 0; i < 20; ++i) { e[i] = __expf(p[i] - m); s += e[i]; }
    const float inv = 1.0f / s;
    #pragma unroll
    for (int i = 0; i < 20; ++i) p[i] = e[i] * inv;
}

// ---------------------------------------------------------------------------
// MSDA bilinear sampling. Thread = (b, q, h, d); 16 consecutive lanes cover
// the 16 channels of one head -> each corner gather is a coalesced 64B read.
//   v: (B, LTOT, 128), off: (Q, 320), aw: (Q, 160) normalized, samp: (B,Q,128)
// ---------------------------------------------------------------------------
__global__ void msda_sample_kernel(const float* __restrict__ v, const float* __restrict__ off,
                                   const float* __restrict__ aw, float* __restrict__ samp)
{
    const int t = blockIdx.x * blockDim.x + threadIdx.x;
    if (t >= BATCH * QTOT * NH * 16) return;
    const int d  = t & 15;
    const int h  = (t >> 4) & 7;
    const int qi = (t >> 7) & (QTOT - 1);
    const int b  = t >> 21;

    const float refx = (float)(qi & (QD - 1)) * (1.0f / (QD - 1));
    const float refy = (float)(qi >> 7)       * (1.0f / (QD - 1));
    const float* __restrict__ offp = off + (size_t)qi * 320 + h * 40;
    const float* __restrict__ awp  = aw  + (size_t)qi * 160 + h * 20;
    const size_t vbase = (size_t)b * LTOT * EMBED + h * 16 + d;

    float acc = 0.0f;
    int start = 0;
    #pragma unroll
    for (int lvl = 0; lvl < NL; ++lvl) {
        const int Wl = 128 >> lvl;
        const int Hl = Wl;
        const float invW = 1.0f / (float)Wl;
        #pragma unroll
        for (int p = 0; p < NP; ++p) {
            const float lx = refx + offp[(lvl * NP + p) * 2]     * invW;
            const float ly = refy + offp[(lvl * NP + p) * 2 + 1] * invW;
            const float px = lx * Wl - 0.5f;
            const float py = ly * Hl - 0.5f;
            const float fx = floorf(px), fy = floorf(py);
            const float tx = px - fx, ty = py - fy;
            const int x0 = (int)fx, y0 = (int)fy;
            float sval = 0.0f;
            auto corner = [&](int xi, int yi, float wc) {
                if (xi >= 0 && xi < Wl && yi >= 0 && yi < Hl)
                    sval += wc * v[vbase + (size_t)(start + yi * Wl + xi) * EMBED];
            };
            corner(x0,     y0,     (1.0f - tx) * (1.0f - ty));
            corner(x0 + 1, y0,     tx * (1.0f - ty));
            corner(x0,     y0 + 1, (1.0f - tx) * ty);
            corner(x0 + 1, y0 + 1, tx * ty);
            acc += awp[lvl * NP + p] * sval;
        }
        start += Wl * Hl;
    }
    samp[((size_t)b * QTOT + qi) * EMBED + h * 16 + d] = acc;
}

// ---------------------------------------------------------------------------
// Final: LayerNorm(h2 row) * gamma + beta, residual add with fused,
// transposed write (B,Q,C) -> (B,C,QD,QD). One 128-thread block per row.
// ---------------------------------------------------------------------------
__global__ __launch_bounds__(128) void ln_residual_out_kernel(
    const float* __restrict__ h2, const float* __restrict__ fused,
    const float* __restrict__ gamma, const float* __restrict__ beta,
    float* __restrict__ out)
{
    const int row = blockIdx.x;            // b*Q + q
    const int c = threadIdx.x;
    const float v = h2[(size_t)row * EMBED + c];
    __shared__ float sh0[128], sh1[128];
    sh0[c] = v; sh1[c] = v * v;
    __syncthreads();
    for (int off = 64; off > 0; off >>= 1) {
        if (c < off) { sh0[c] += sh0[c + off]; sh1[c] += sh1[c + off]; }
        __syncthreads();
    }
    __shared__ float mu_s, rs_s;
    if (c == 0) {
        const float mu = sh0[0] * (1.0f / EMBED);
        const float var = sh1[0] * (1.0f / EMBED) - mu * mu;
        mu_s = mu; rs_s = rsqrtf(var + 1e-5f);
    }
    __syncthreads();
    const float hn = (v - mu_s) * rs_s * gamma[c] + beta[c];
    const float o = fused[(size_t)row * EMBED + c] + hn;
    const int b = row >> 14, q = row & (QTOT - 1);
    out[((size_t)b * EMBED + c) * QTOT + q] = o;
}

// ---------------------------------------------------------------------------
// Host orchestration
// ---------------------------------------------------------------------------
static inline void launch_gemm(const float* X, const void* W, const void* bias, float* Y,
                               int rows, int K, int N, int relu, int scaleMode,
                               const float* alpha, int accum, hipStream_t st)
{
    const int tiles = (rows / 32) * (N / 32);    // one wave per 32x32 macro tile
    gemm_wmma_f32<<<cdiv(tiles, 8), 256, 0, st>>>(
        X, (const float*)W, (const float*)bias, Y, rows, K, N,
        relu, scaleMode, alpha, accum);
}

extern "C" void kernel_launch(void* const* d_in, const int* in_sizes, int n_in,
                              void* d_out, int out_size, void* d_ws, size_t ws_size,
                              hipStream_t stream)
{
    (void)in_sizes; (void)n_in; (void)out_size; (void)ws_size;
    float* ws = (float*)d_ws;
    const float* alpha = (const float*)d_in[69];

    // workspace regions (floats); FFN hidden/h2 overlay the dead MSDA scratch
    const size_t FLAT_OFF  = 0;                               // 43648*128
    const size_t V_OFF     = FLAT_OFF  + (size_t)BATCH * LTOT * EMBED;   //  5,586,944
    const size_t Q_OFF     = V_OFF     + (size_t)BATCH * LTOT * EMBED;   // 11,173,888
    const size_t OFF_OFF   = Q_OFF     + (size_t)QTOT * EMBED;           // 13,271,040
    const size_t AW_OFF    = OFF_OFF   + (size_t)QTOT * 320;             // 18,513,920
    const size_t SAMP_OFF  = AW_OFF    + (size_t)QTOT * 160;             // 21,135,360
    const size_t FUSED_OFF = SAMP_OFF  + (size_t)BATCH * QTOT * EMBED;   // 25,329,664
    const size_t STATS_OFF = FUSED_OFF + (size_t)BATCH * QTOT * EMBED;   // 29,523,968
    const size_t H_OFF  = 0;                       // (32768x512) overlays flat..aw
    const size_t H2_OFF = (size_t)BATCH * QTOT * HIDDEN;   // after H, before SAMP

    const int CH_[5]  = {16, 24, 40, 112, 320};
    const int HW_[5]  = {16384, 4096, 1024, 256, 64};
    const int LOG_[5] = {14, 12, 10, 8, 6};
    const int ST_[5]  = {0, 16384, 20480, 21504, 21760};

    // q = learnable_Q + pos  (batch-invariant, computed once on Q rows)
    add_vec_kernel<<<cdiv(QTOT * EMBED, 256), 256, 0, stream>>>(
        (const float*)d_in[67], (const float*)d_in[68], ws + Q_OFF, QTOT * EMBED);

    for (int s = 0; s < 2; ++s) {
        const int fb = s ? 5 : 0;      // feature levels base index
        const int pb = s ? 31 : 11;    // proj params base index
        const int mb = s ? 59 : 51;    // msda params base index

        // prepare: conv1x1 -> GroupNorm -> flat (B, LTOT, 128)
        for (int l = 0; l < 5; ++l) {
            const int tot = BATCH * HW_[l] * EMBED;
            conv1x1_kernel<<<cdiv(tot, 256), 256, 0, stream>>>(
                (const float*)d_in[fb + l], (const float*)d_in[pb + l * 4 + 0],
                (const float*)d_in[pb + l * 4 + 1], ws + FLAT_OFF,
                CH_[l], HW_[l], LOG_[l], ST_[l]);
            gn_stats_kernel<<<BATCH * 32, 256, 0, stream>>>(
                ws + FLAT_OFF, ws + STATS_OFF, HW_[l], ST_[l]);
            gn_apply_kernel<<<cdiv(tot, 256), 256, 0, stream>>>(
                ws + FLAT_OFF, ws + STATS_OFF,
                (const float*)d_in[pb + l * 4 + 2], (const float*)d_in[pb + l * 4 + 3],
                HW_[l], LOG_[l], ST_[l]);
        }

        // value projection: (B*LTOT,128) @ Wv
        launch_gemm(ws + FLAT_OFF, d_in[mb + 0], d_in[mb + 1], ws + V_OFF,
                    BATCH * LTOT, EMBED, EMBED, 0, 0, alpha, 0, stream);
        // sampling offsets: (Q,128) @ Wo -> (Q,320)
        launch_gemm(ws + Q_OFF, d_in[mb + 2], d_in[mb + 3], ws + OFF_OFF,
                    QTOT, EMBED, NH * NL * NP * 2, 0, 0, alpha, 0, stream);
        // attention logits: (Q,128) @ Wa -> (Q,160), then softmax over 20
        launch_gemm(ws + Q_OFF, d_in[mb + 4], d_in[mb + 5], ws + AW_OFF,
                    QTOT, EMBED, NH * NL * NP, 0, 0, alpha, 0, stream);
        softmax20_kernel<<<cdiv(QTOT * NH, 256), 256, 0, stream>>>(ws + AW_OFF);

        // deformable bilinear sampling -> (B,Q,128)
        msda_sample_kernel<<<cdiv(BATCH * QTOT * EMBED, 256), 256, 0, stream>>>(
            ws + V_OFF, ws + OFF_OFF, ws + AW_OFF, ws + SAMP_OFF);

        // output projection, fused with alpha-blend: sat writes, osm accumulates
        launch_gemm(ws + SAMP_OFF, d_in[mb + 6], d_in[mb + 7], ws + FUSED_OFF,
                    BATCH * QTOT, EMBED, EMBED, 0, s ? 2 : 1, alpha, s ? 1 : 0, stream);
    }

    // FFN: relu(fused @ W1 + b1) @ W2 + b2, then LN + residual + transpose out
    launch_gemm(ws + FUSED_OFF, d_in[70], d_in[71], ws + H_OFF,
                BATCH * QTOT, EMBED, HIDDEN, 1, 0, alpha, 0, stream);
    launch_gemm(ws + H_OFF, d_in[72], d_in[73], ws + H2_OFF,
                BATCH * QTOT, HIDDEN, EMBED, 0, 0, alpha, 0, stream);
    ln_residual_out_kernel<<<BATCH * QTOT, 128, 0, stream>>>(
        ws + H2_OFF, ws + FUSED_OFF, (const float*)d_in[74], (const float*)d_in[75],
        (float*)d_out);
}